// HGNNEncoder_40913858461771
// MI455X (gfx1250) — compile-verified
//
#include <hip/hip_runtime.h>

#define TT  8
#define NN  50000
#define EE  5000
#define NNZP 100000
#define FF  128
#define DD  128

typedef float v2f __attribute__((ext_vector_type(2)));
typedef float v8f __attribute__((ext_vector_type(8)));

// ---------------- generic fill ----------------
__global__ void hgnn_fill_kernel(float* __restrict__ p, float v, int n) {
    for (int i = blockIdx.x * blockDim.x + threadIdx.x; i < n; i += gridDim.x * blockDim.x)
        p[i] = v;
}

// ---------------- degree counting ----------------
__global__ void hgnn_degree_kernel(const int* __restrict__ pn, const int* __restrict__ pe,
                                   float* __restrict__ dv, float* __restrict__ de) {
    int i = blockIdx.x * blockDim.x + threadIdx.x;
    if (i < NNZP) {
        atomicAdd(&dv[pn[i]], 1.0f);
        atomicAdd(&de[pe[i]], 1.0f);
    }
}

// mode 0: d := d>0 ? rsqrt(d) : 0   (node);  mode 1: d := d>0 ? 1/d : 0  (edge)
__global__ void hgnn_finalize_deg_kernel(float* __restrict__ d, int n, int mode) {
    int i = blockIdx.x * blockDim.x + threadIdx.x;
    if (i < n) {
        float v = d[i];
        d[i] = (v > 0.0f) ? (mode == 0 ? rsqrtf(v) : 1.0f / v) : 0.0f;
    }
}

// ---------------- WMMA fp32 GEMM: out = (X @ W + bias) * rowscale ----------------
// X: [nrows,128] row-major, W: [128,128] row-major. One block = 16 rows x 128 cols.
// 8 waves; wave w owns column tile w*16. Uses V_WMMA_F32_16X16X4_F32, K stepped by 4.
__global__ __launch_bounds__(256) void hgnn_gemm_bias_scale_kernel(
    const float* __restrict__ X, const float* __restrict__ W,
    const float* __restrict__ bias, const float* __restrict__ rowscale,
    float* __restrict__ out, int nrows) {
    __shared__ float lds_x[16][132];   // 132-float pitch: 132 % 64 = 4 -> conflict-free rows

    const int rowbase = blockIdx.x * 16;
    const int tid = threadIdx.x;

    // cooperative 16x128 tile load (512 float4 by 256 threads)
    #pragma unroll
    for (int it = 0; it < 2; ++it) {
        int idx = tid + it * 256;        // 0..511
        int r   = idx >> 5;              // 0..15
        int c4  = (idx & 31) * 4;        // 0..124
        float4 v = make_float4(0.f, 0.f, 0.f, 0.f);
        int row = rowbase + r;
        if (row < nrows) v = *reinterpret_cast<const float4*>(&X[(size_t)row * FF + c4]);
        lds_x[r][c4 + 0] = v.x; lds_x[r][c4 + 1] = v.y;
        lds_x[r][c4 + 2] = v.z; lds_x[r][c4 + 3] = v.w;
    }
    __syncthreads();

    const int wave    = tid >> 5;        // 0..7  -> column tile
    const int lane    = tid & 31;
    const int lm      = lane & 15;       // M for A-frag, N for B-frag
    const int kb      = (lane >> 4) * 2; // lane half selects K 0..1 vs 2..3
    const int colbase = wave * 16;
    const int col     = colbase + lm;

    v8f acc = {};
    #pragma unroll
    for (int k = 0; k < FF; k += 4) {
        v2f a, b;
        a[0] = lds_x[lm][k + kb + 0];               // A: 16x4 tile of X
        a[1] = lds_x[lm][k + kb + 1];
        b[0] = W[(size_t)(k + kb + 0) * DD + col];  // B: 4x16 tile of W
        b[1] = W[(size_t)(k + kb + 1) * DD + col];
        acc = __builtin_amdgcn_wmma_f32_16x16x4_f32(
            /*neg_a=*/false, a, /*neg_b=*/false, b,
            /*c_mod=*/(short)0, acc, /*reuse_a=*/false, /*reuse_b=*/false);
    }

    // C/D layout: VGPR r -> M = r + 8*(lane>=16), N = lane&15
    const float bv   = bias[col];
    const int mbase  = rowbase + (lane >> 4) * 8;
    #pragma unroll
    for (int r = 0; r < 8; ++r) {
        int row = mbase + r;
        if (row < nrows)
            out[(size_t)row * DD + col] = (acc[r] + bv) * rowscale[row];
    }
}

// ---------------- node -> hyperedge atomic scatter: ef[pe[i]] += xs[pn[i]] ----------------
__global__ void hgnn_scatter_n2e_kernel(const float* __restrict__ xs,
                                        const int* __restrict__ pn, const int* __restrict__ pe,
                                        float* __restrict__ ef) {
    int pair = blockIdx.x;
    int d = threadIdx.x;
    int n = pn[pair], e = pe[pair];
    atomicAdd(&ef[(size_t)e * DD + d], xs[(size_t)n * DD + d]);
}

// ---------------- hyperedge -> node atomic scatter: out[pn[i]] += ef[pe[i]] * de_inv[pe[i]] ----
__global__ void hgnn_scatter_e2n_kernel(const float* __restrict__ ef, const float* __restrict__ de_inv,
                                        const int* __restrict__ pn, const int* __restrict__ pe,
                                        float* __restrict__ outb) {
    int pair = blockIdx.x;
    int d = threadIdx.x;
    int n = pn[pair], e = pe[pair];
    atomicAdd(&outb[(size_t)n * DD + d], ef[(size_t)e * DD + d] * de_inv[e]);
}

// ---------------- h := relu(h * dv_isqrt[row]) ----------------
__global__ void hgnn_scale_relu_kernel(float* __restrict__ h, const float* __restrict__ dvi, int n_elems) {
    int i = blockIdx.x * blockDim.x + threadIdx.x;
    if (i < n_elems) {
        float v = h[i] * dvi[i >> 7];   // DD == 128
        h[i] = v > 0.0f ? v : 0.0f;
    }
}

// ---------------- mean over nodes -> out[d] ----------------
__global__ void hgnn_mean_pool_kernel(const float* __restrict__ h, float* __restrict__ out) {
    int d = threadIdx.x;                // 128
    int start = blockIdx.x * 256;
    float s = 0.0f;
    #pragma unroll 4
    for (int r = 0; r < 256; ++r) {
        int n = start + r;
        if (n < NN) s += h[(size_t)n * DD + d];
    }
    atomicAdd(&out[d], s * (1.0f / (float)NN));
}

extern "C" void kernel_launch(void* const* d_in, const int* in_sizes, int n_in,
                              void* d_out, int out_size, void* d_ws, size_t ws_size,
                              hipStream_t stream) {
    const float* x      = (const float*)d_in[0];   // [T,N,F]
    const float* W0     = (const float*)d_in[1];   // [F,D]
    const float* b0     = (const float*)d_in[2];   // [D]
    const float* W1     = (const float*)d_in[3];   // [D,D]
    const float* b1     = (const float*)d_in[4];   // [D]
    const int*   pn_all = (const int*)d_in[5];     // [T,NNZ]
    const int*   pe_all = (const int*)d_in[6];     // [T,NNZ]
    float* out = (float*)d_out;                    // [1,T,D] = 1024 floats

    // workspace layout (floats): bufA[N*D] bufB[N*D] ef[E*D] dv[N] de[E]  (~54 MB)
    float* ws   = (float*)d_ws;
    float* bufA = ws;
    float* bufB = bufA + (size_t)NN * DD;
    float* ef   = bufB + (size_t)NN * DD;
    float* dv   = ef   + (size_t)EE * DD;
    float* de   = dv   + NN;

    hgnn_fill_kernel<<<4, 256, 0, stream>>>(out, 0.0f, TT * DD);

    for (int t = 0; t < TT; ++t) {
        const float* xt = x + (size_t)t * NN * FF;
        const int* pn = pn_all + (size_t)t * NNZP;
        const int* pe = pe_all + (size_t)t * NNZP;

        // degrees -> dv_isqrt, de_inv (in place)
        hgnn_fill_kernel<<<128, 256, 0, stream>>>(dv, 0.0f, NN);
        hgnn_fill_kernel<<<32, 256, 0, stream>>>(de, 0.0f, EE);
        hgnn_degree_kernel<<<(NNZP + 255) / 256, 256, 0, stream>>>(pn, pe, dv, de);
        hgnn_finalize_deg_kernel<<<(NN + 255) / 256, 256, 0, stream>>>(dv, NN, 0);
        hgnn_finalize_deg_kernel<<<(EE + 255) / 256, 256, 0, stream>>>(de, EE, 1);

        // ---- layer 0: bufA = (x_t @ W0 + b0) * dv_isqrt ----
        hgnn_gemm_bias_scale_kernel<<<(NN + 15) / 16, 256, 0, stream>>>(xt, W0, b0, dv, bufA, NN);
        hgnn_fill_kernel<<<256, 256, 0, stream>>>(ef, 0.0f, EE * DD);
        hgnn_scatter_n2e_kernel<<<NNZP, DD, 0, stream>>>(bufA, pn, pe, ef);
        hgnn_fill_kernel<<<1024, 256, 0, stream>>>(bufB, 0.0f, NN * DD);
        hgnn_scatter_e2n_kernel<<<NNZP, DD, 0, stream>>>(ef, de, pn, pe, bufB);
        hgnn_scale_relu_kernel<<<(NN * DD + 255) / 256, 256, 0, stream>>>(bufB, dv, NN * DD);

        // ---- layer 1: bufA = (bufB @ W1 + b1) * dv_isqrt ----
        hgnn_gemm_bias_scale_kernel<<<(NN + 15) / 16, 256, 0, stream>>>(bufB, W1, b1, dv, bufA, NN);
        hgnn_fill_kernel<<<256, 256, 0, stream>>>(ef, 0.0f, EE * DD);
        hgnn_scatter_n2e_kernel<<<NNZP, DD, 0, stream>>>(bufA, pn, pe, ef);
        hgnn_fill_kernel<<<1024, 256, 0, stream>>>(bufB, 0.0f, NN * DD);
        hgnn_scatter_e2n_kernel<<<NNZP, DD, 0, stream>>>(ef, de, pn, pe, bufB);
        hgnn_scale_relu_kernel<<<(NN * DD + 255) / 256, 256, 0, stream>>>(bufB, dv, NN * DD);

        // ---- mean pool over nodes ----
        hgnn_mean_pool_kernel<<<(NN + 255) / 256, DD, 0, stream>>>(bufB, out + t * DD);
    }
}